// SMPLModel_47519518163082
// MI455X (gfx1250) — compile-verified
//
#include <hip/hip_runtime.h>
#include <hip/hip_bf16.h>
#include <math.h>

// ---- problem constants ----
#define BB 512
#define VV 6890
#define NJ 24
#define NB 10

typedef __attribute__((ext_vector_type(2))) float v2f;
typedef __attribute__((ext_vector_type(8))) float v8f;

__device__ __constant__ int c_parents[NJ] = {
    -1, 0, 0, 0, 1, 2, 3, 4, 5, 6, 7, 8, 9, 9, 9, 12, 13, 14, 16, 17, 18, 19, 20, 21};

// ------------------------------------------------------------------
// Kernel 1: v_shaped[b,e] = (v_template[e] + sum_l shapedirs[e,l]*betas[b,1+l]) * betas[b,0]
//   e = flat index over V*3.  Bandwidth bound (42 MB write).
// ------------------------------------------------------------------
__global__ void shape_blend_kernel(const float* __restrict__ betas,
                                   const float* __restrict__ shapedirs,
                                   const float* __restrict__ v_template,
                                   float* __restrict__ vs) {
    int b = blockIdx.y;
    int e = blockIdx.x * blockDim.x + threadIdx.x;
    if (e >= VV * 3) return;
    const float* bb = betas + b * (NB + 1);       // uniform -> scalar loads
    float scale = bb[0];
    const float* sd = shapedirs + (size_t)e * NB;
    float acc = v_template[e];
#pragma unroll
    for (int l = 0; l < NB; ++l) acc += sd[l] * bb[1 + l];
    vs[(size_t)b * VV * 3 + e] = acc * scale;
}

// ------------------------------------------------------------------
// Kernel 2: J[b,j,k] = sum_v Jr[j,v] * vs[b,v,k]
//   one block (256 thr) per batch; 72 register accumulators per thread,
//   wave32 shuffle reduce, then ds_add_f32 cross-wave reduce.
// ------------------------------------------------------------------
__global__ void jreg_kernel(const float* __restrict__ vs,
                            const float* __restrict__ Jr,
                            float* __restrict__ Jout) {
    int b = blockIdx.x;
    int t = threadIdx.x;
    float acc[NJ][3];
#pragma unroll
    for (int j = 0; j < NJ; ++j) {
        acc[j][0] = 0.f; acc[j][1] = 0.f; acc[j][2] = 0.f;
    }
    const float* vsb = vs + (size_t)b * VV * 3;
    for (int v = t; v < VV; v += 256) {
        float x = vsb[v * 3 + 0];
        float y = vsb[v * 3 + 1];
        float z = vsb[v * 3 + 2];
#pragma unroll
        for (int j = 0; j < NJ; ++j) {
            float w = Jr[j * VV + v];
            acc[j][0] += w * x; acc[j][1] += w * y; acc[j][2] += w * z;
        }
    }
    // wave32 reduce
#pragma unroll
    for (int j = 0; j < NJ; ++j) {
#pragma unroll
        for (int k = 0; k < 3; ++k) {
            float s = acc[j][k];
            for (int off = 16; off > 0; off >>= 1) s += __shfl_xor(s, off, 32);
            acc[j][k] = s;
        }
    }
    __shared__ float red[NJ * 3];
    if (t < NJ * 3) red[t] = 0.f;
    __syncthreads();
    if ((t & 31) == 0) {
#pragma unroll
        for (int j = 0; j < NJ; ++j)
            for (int k = 0; k < 3; ++k)
                atomicAdd(&red[j * 3 + k], acc[j][k]);   // ds_add_f32
    }
    __syncthreads();
    if (t < NJ * 3) Jout[(size_t)b * NJ * 3 + t] = red[t];
}

// ------------------------------------------------------------------
// Kernel 3: Rodrigues + kinematic chain + relative transforms A.
//   one wave per batch; lane j does rodrigues for joint j; 12 lanes
//   compose each 3x4 affine link in LDS.  Outputs A[b,24,12] (ws) and
//   J_transformed[b,24,3] (d_out tail).
// ------------------------------------------------------------------
__global__ void chain_kernel(const float* __restrict__ global_orient,
                             const float* __restrict__ body_pose,
                             const float* __restrict__ Jws,
                             float* __restrict__ Aws,
                             float* __restrict__ Jtr_out) {
    int b = blockIdx.x;
    int lane = threadIdx.x;
    __shared__ float Tl[NJ][12];
    __shared__ float ch[NJ][12];
    __shared__ float Jl[NJ][3];

    for (int i = lane; i < NJ * 3; i += 32) Jl[i / 3][i % 3] = Jws[(size_t)b * NJ * 3 + i];
    __syncthreads();

    if (lane < NJ) {
        float rx, ry, rz;
        if (lane == 0) {
            rx = global_orient[b * 3 + 0];
            ry = global_orient[b * 3 + 1];
            rz = global_orient[b * 3 + 2];
        } else {
            const float* p = body_pose + (size_t)b * 3 * (NJ - 1) + (lane - 1) * 3;
            rx = p[0]; ry = p[1]; rz = p[2];
        }
        // reference: angle = ||r + 1e-8||, dir = r / angle
        float ax = rx + 1e-8f, ay = ry + 1e-8f, az = rz + 1e-8f;
        float angle = sqrtf(ax * ax + ay * ay + az * az);
        float inv = 1.0f / angle;
        float ux = rx * inv, uy = ry * inv, uz = rz * inv;
        float s = sinf(angle), c = cosf(angle);
        float omc = 1.0f - c;
        float R00 = 1.0f - omc * (uy * uy + uz * uz);
        float R01 = -s * uz + omc * ux * uy;
        float R02 =  s * uy + omc * ux * uz;
        float R10 =  s * uz + omc * ux * uy;
        float R11 = 1.0f - omc * (ux * ux + uz * uz);
        float R12 = -s * ux + omc * uy * uz;
        float R20 = -s * uy + omc * ux * uz;
        float R21 =  s * ux + omc * uy * uz;
        float R22 = 1.0f - omc * (ux * ux + uy * uy);
        int p = c_parents[lane];
        float tx, ty, tz;
        if (lane == 0) { tx = Jl[0][0]; ty = Jl[0][1]; tz = Jl[0][2]; }
        else {
            tx = Jl[lane][0] - Jl[p][0];
            ty = Jl[lane][1] - Jl[p][1];
            tz = Jl[lane][2] - Jl[p][2];
        }
        Tl[lane][0] = R00; Tl[lane][1] = R01; Tl[lane][2]  = R02; Tl[lane][3]  = tx;
        Tl[lane][4] = R10; Tl[lane][5] = R11; Tl[lane][6]  = R12; Tl[lane][7]  = ty;
        Tl[lane][8] = R20; Tl[lane][9] = R21; Tl[lane][10] = R22; Tl[lane][11] = tz;
    }
    __syncthreads();

    if (lane < 12) ch[0][lane] = Tl[0][lane];
    __syncthreads();
    for (int i = 1; i < NJ; ++i) {
        int p = c_parents[i];
        if (lane < 12) {
            int r = lane >> 2, cc = lane & 3;
            float v = ch[p][r * 4 + 0] * Tl[i][0 * 4 + cc]
                    + ch[p][r * 4 + 1] * Tl[i][1 * 4 + cc]
                    + ch[p][r * 4 + 2] * Tl[i][2 * 4 + cc];
            if (cc == 3) v += ch[p][r * 4 + 3];
            ch[i][lane] = v;
        }
        __syncthreads();
    }

    for (int idx = lane; idx < NJ * 3; idx += 32) {
        int j = idx / 3, k = idx % 3;
        Jtr_out[(size_t)b * NJ * 3 + idx] = ch[j][k * 4 + 3];
    }
    for (int idx = lane; idx < NJ * 12; idx += 32) {
        int j = idx / 12, e = idx % 12, r = e >> 2, cc = e & 3;
        float v = ch[j][e];
        if (cc == 3)
            v -= ch[j][r * 4 + 0] * Jl[j][0]
               + ch[j][r * 4 + 1] * Jl[j][1]
               + ch[j][r * 4 + 2] * Jl[j][2];
        Aws[(size_t)b * NJ * 12 + idx] = v;
    }
}

// ------------------------------------------------------------------
// Kernel 4 (WMMA): per batch, Tmat = W[V,24] x A[b][24,12] via
// V_WMMA_F32_16X16X4_F32 (M=16 verts, K=24 = 6 chained WMMAs, N=12/16),
// then verts = Tmat[:3,:3]*v_shaped + Tmat[:,3] + transl.
//   8 waves/block, one 16-vertex tile per wave.
// f32 fragment layout (ISA 7.12.2): K = vgprIdx + 2*(lane>>4), M/N = lane&15.
// ------------------------------------------------------------------
#define TPB_TILES 8
__global__ void skin_kernel(const float* __restrict__ W,
                            const float* __restrict__ Aws,
                            const float* __restrict__ vs,
                            const float* __restrict__ transl,
                            float* __restrict__ verts) {
    int b = blockIdx.y;
    int wave = threadIdx.x >> 5;
    int lane = threadIdx.x & 31;
    int half = lane >> 4;
    int lh = lane & 15;
    int tile = blockIdx.x * TPB_TILES + wave;

    __shared__ float Tsh[TPB_TILES][16][13];

    const float* Ab = Aws + (size_t)b * NJ * 12;

    // B fragments (joint transforms), uniform per wave; zero-pad cols 12..15
    v2f bfr[6];
#pragma unroll
    for (int kc = 0; kc < 6; ++kc) {
        int k0 = 4 * kc + 2 * half;
        float bx = 0.f, by = 0.f;
        if (lh < 12) {
            bx = Ab[(k0 + 0) * 12 + lh];
            by = Ab[(k0 + 1) * 12 + lh];
        }
        bfr[kc].x = bx; bfr[kc].y = by;
    }

    int m = tile * 16 + lh;
    bool mvalid = (m < VV);
    const float* Wrow = W + (size_t)(mvalid ? m : 0) * NJ;

    v8f c = {0.f, 0.f, 0.f, 0.f, 0.f, 0.f, 0.f, 0.f};
#pragma unroll
    for (int kc = 0; kc < 6; ++kc) {
        v2f a;
        float ax = Wrow[4 * kc + 2 * half + 0];   // contiguous pair -> b64 load
        float ay = Wrow[4 * kc + 2 * half + 1];
        a.x = mvalid ? ax : 0.f;
        a.y = mvalid ? ay : 0.f;
        c = __builtin_amdgcn_wmma_f32_16x16x4_f32(
                /*neg_a=*/false, a, /*neg_b=*/false, bfr[kc],
                /*c_mod=*/(short)0, c, /*reuse_a=*/false, /*reuse_b=*/false);
    }

    // D layout: lane = column N, vgpr i = row M (halves split at M=8)
    if (lh < 12) {
#pragma unroll
        for (int i = 0; i < 8; ++i)
            Tsh[wave][half * 8 + i][lh] = c[i];
    }
    // same-wave LDS write -> cross-lane read: drain the DS counter
    asm volatile("s_wait_dscnt 0" ::: "memory");

    if (half == 0) {
        int mv = tile * 16 + lh;
        if (mv < VV) {
            float t0[12];
#pragma unroll
            for (int e = 0; e < 12; ++e) t0[e] = Tsh[wave][lh][e];
            const float* pv = vs + ((size_t)b * VV + mv) * 3;
            float x = pv[0], y = pv[1], z = pv[2];
            float tx = transl[b * 3 + 0];
            float ty = transl[b * 3 + 1];
            float tz = transl[b * 3 + 2];
            float ox = t0[0] * x + t0[1] * y + t0[2]  * z + t0[3]  + tx;
            float oy = t0[4] * x + t0[5] * y + t0[6]  * z + t0[7]  + ty;
            float oz = t0[8] * x + t0[9] * y + t0[10] * z + t0[11] + tz;
            float* po = verts + ((size_t)b * VV + mv) * 3;
            po[0] = ox; po[1] = oy; po[2] = oz;
        }
    }
}

// ------------------------------------------------------------------
extern "C" void kernel_launch(void* const* d_in, const int* in_sizes, int n_in,
                              void* d_out, int out_size, void* d_ws, size_t ws_size,
                              hipStream_t stream) {
    const float* betas         = (const float*)d_in[0];   // [B, NB+1]
    const float* body_pose     = (const float*)d_in[1];   // [B, 69]
    const float* global_orient = (const float*)d_in[2];   // [B, 3]
    const float* transl        = (const float*)d_in[3];   // [B, 3]
    const float* shapedirs     = (const float*)d_in[4];   // [V, 3, NB]
    const float* v_template    = (const float*)d_in[5];   // [V, 3]
    const float* Jreg          = (const float*)d_in[6];   // [NJ, V]
    const float* Wlbs          = (const float*)d_in[7];   // [V, NJ]

    float* out   = (float*)d_out;
    float* verts = out;                                   // [B, V, 3]
    float* Jtr   = out + (size_t)BB * VV * 3;             // [B, NJ, 3]

    float* ws  = (float*)d_ws;
    float* vsb = ws;                                      // v_shaped [B, V, 3]
    float* Jws = vsb + (size_t)BB * VV * 3;               // J        [B, NJ, 3]
    float* Aws = Jws + (size_t)BB * NJ * 3;               // A        [B, NJ, 12]

    // 1) shape blend
    shape_blend_kernel<<<dim3((VV * 3 + 255) / 256, BB), 256, 0, stream>>>(
        betas, shapedirs, v_template, vsb);
    // 2) joint regression
    jreg_kernel<<<dim3(BB), 256, 0, stream>>>(vsb, Jreg, Jws);
    // 3) rodrigues + kinematic chain + A
    chain_kernel<<<dim3(BB), 32, 0, stream>>>(global_orient, body_pose, Jws, Aws, Jtr);
    // 4) WMMA skinning
    int ntiles = (VV + 15) / 16;                          // 431
    skin_kernel<<<dim3((ntiles + TPB_TILES - 1) / TPB_TILES, BB), 256, 0, stream>>>(
        Wlbs, Aws, vsb, transl, verts);
}